// piotr_event_cam_spatial_82016695485087
// MI455X (gfx1250) — compile-verified
//
#include <hip/hip_runtime.h>

#define NCLS   10
#define FANIN  4096
#define BATCH  64
#define TSTEPS 128

typedef __attribute__((ext_vector_type(2))) float v2f;
typedef __attribute__((ext_vector_type(4))) float v4f;
typedef __attribute__((ext_vector_type(8))) float v8f;

// ---------------------------------------------------------------------------
// Kernel 1: stream the 134 MB input once, produce S[b,k] = sum_t sign(dx).
// 65536 threads, each owns 4 consecutive k and walks 128 timesteps.
// Loads are 16B, fully coalesced (4 KB contiguous per block per timestep),
// non-temporal since every byte is touched exactly once.
// ---------------------------------------------------------------------------
__global__ __launch_bounds__(256) void sign_sum_kernel(const float* __restrict__ x,
                                                       float* __restrict__ S) {
  const int gid = blockIdx.x * 256 + threadIdx.x;   // 0 .. 65535
  const int b   = gid >> 10;                        // 1024 float4-groups per batch row
  const int k4  = (gid & 1023) << 2;
  const v4f* __restrict__ xp = (const v4f*)(x + (size_t)b * TSTEPS * FANIN + k4);
  const int stride4 = FANIN / 4;                    // float4 stride between timesteps

  v4f prev = __builtin_nontemporal_load(xp);
  v4f s;  s.x = 0.f; s.y = 0.f; s.z = 0.f; s.w = 0.f;

  #pragma unroll 4
  for (int t = 1; t < TSTEPS; ++t) {
    v4f cur = __builtin_nontemporal_load(xp + t * stride4);
    s.x += (cur.x > prev.x ? 1.f : 0.f) - (cur.x < prev.x ? 1.f : 0.f);
    s.y += (cur.y > prev.y ? 1.f : 0.f) - (cur.y < prev.y ? 1.f : 0.f);
    s.z += (cur.z > prev.z ? 1.f : 0.f) - (cur.z < prev.z ? 1.f : 0.f);
    s.w += (cur.w > prev.w ? 1.f : 0.f) - (cur.w < prev.w ? 1.f : 0.f);
    prev = cur;
  }
  *(v4f*)(S + (size_t)b * FANIN + k4) = s;
}

// ---------------------------------------------------------------------------
// Kernel 2: out(64x10) = (S(64x4096) @ Wfc^T(4096x10->pad16) + 127*bfc) / 128
// One wave per 16x16 output tile; K=4096 reduced with 1024 chained
// V_WMMA_F32_16X16X4_F32 ops (full f32 precision, matching the reference).
//
// f32 WMMA operand layout (ISA 7.12.2):
//   A (16x4):  M = lane&15 (both halves);  VGPR0/1 = K = k0+{0,1} (lanes 0-15)
//                                                   = k0+{2,3} (lanes 16-31)
//   B (4x16):  N = lane&15, mirrored K split across lane halves.
//   C/D      : VGPR r -> M = r + 8*(lane>=16), N = lane&15.
// EXEC must be all ones at every WMMA: the N>=10 zero-padding is branch-free.
// ---------------------------------------------------------------------------
__global__ __launch_bounds__(32) void wmma_fc_kernel(const float* __restrict__ S,
                                                     const float* __restrict__ Wfc,
                                                     const float* __restrict__ bfc,
                                                     float* __restrict__ out) {
  const int lane = threadIdx.x;          // one wave32 per block
  const int tile = blockIdx.x;           // 0..3 -> output rows 16*tile..
  const int half = lane >> 4;            // 0: K+{0,1}, 1: K+{2,3}
  const int n    = lane & 15;

  const float* __restrict__ arow = S + (size_t)(tile * 16 + n) * FANIN;
  // zero-pad columns n >= NCLS without diverging EXEC
  const float  bmask = (n < NCLS) ? 1.0f : 0.0f;
  const float* __restrict__ brow = Wfc + (size_t)(n < NCLS ? n : 0) * FANIN;

  v8f acc = {};
  #pragma unroll 4
  for (int k0 = 0; k0 < FANIN; k0 += 4) {
    const int kk = k0 + (half << 1);     // even -> 8B-aligned v2f loads
    v2f a = *(const v2f*)(arow + kk);
    v2f bv = *(const v2f*)(brow + kk);
    bv.x *= bmask;
    bv.y *= bmask;
    // 8 args: (neg_a, A, neg_b, B, c_mod, C, reuse_a, reuse_b)
    acc = __builtin_amdgcn_wmma_f32_16x16x4_f32(
        /*neg_a=*/false, a, /*neg_b=*/false, bv,
        /*c_mod=*/(short)0, acc, /*reuse_a=*/false, /*reuse_b=*/false);
  }

  if (n < NCLS) {
    const float bias127 = 127.0f * bfc[n];
    #pragma unroll
    for (int r = 0; r < 8; ++r) {
      const int row = tile * 16 + r + half * 8;
      out[row * NCLS + n] = (acc[r] + bias127) * (1.0f / 128.0f);
    }
  }
}

extern "C" void kernel_launch(void* const* d_in, const int* in_sizes, int n_in,
                              void* d_out, int out_size, void* d_ws, size_t ws_size,
                              hipStream_t stream) {
  const float* x   = (const float*)d_in[0];   // (64,128,1,64,64) f32
  const float* Wfc = (const float*)d_in[1];   // (10,4096) f32
  const float* bfc = (const float*)d_in[2];   // (10,) f32
  float* out = (float*)d_out;                 // (64,10) f32
  float* S   = (float*)d_ws;                  // 64*4096 f32 = 1 MB scratch

  sign_sum_kernel<<<(BATCH * FANIN / 4) / 256, 256, 0, stream>>>(x, S);
  wmma_fc_kernel<<<4, 32, 0, stream>>>(S, Wfc, bfc, out);
}